// GCN_encoder_spmm_10969346474302
// MI455X (gfx1250) — compile-verified
//
#include <hip/hip_runtime.h>

#define N_NODES 100000
#define N_FEAT  512
#define HIDDEN  128
#define KB      32            // K-block staged in LDS per iteration
#define PAD     4             // row padding (floats): 144B row stride (16B mult.)
#define ROWS_PER_WG 128       // 8 waves * 16 rows
#define THREADS 256

typedef float v2f __attribute__((ext_vector_type(2)));
typedef float v8f __attribute__((ext_vector_type(8)));

// ---------------------------------------------------------------------------
// CDNA5 async global->LDS copy (16B per lane), ASYNCcnt-tracked.
// LDS byte offset = low 32 bits of the generic pointer (ISA 10.2: flat->LDS
// aperture mapping truncates to addr[31:0]).
// ---------------------------------------------------------------------------
__device__ __forceinline__ void async_cp16(void* lds_dst, const float* g_src) {
  asm volatile("global_load_async_to_lds_b128 %0, %1, off"
               :: "v"((unsigned int)(uintptr_t)lds_dst), "v"(g_src)
               : "memory");
}
__device__ __forceinline__ void wait_async0() {
  asm volatile("s_wait_asynccnt 0x0" ::: "memory");
}

// ---------------------------------------------------------------------------
// Kernel 0: Wt = W^T  (512x128 -> 128x512), one-time 256KB transpose so the
// GEMM's W tile staging becomes a pure strided block copy (async-LDS capable).
// ---------------------------------------------------------------------------
__global__ void transpose_W(const float* __restrict__ W, float* __restrict__ Wt) {
  int idx = blockIdx.x * blockDim.x + threadIdx.x;
  if (idx < N_FEAT * HIDDEN) {
    int k = idx / HIDDEN;
    int n = idx % HIDDEN;
    Wt[n * N_FEAT + k] = W[idx];
  }
}

// ---------------------------------------------------------------------------
// Kernel 1: h = x @ W via V_WMMA_F32_16X16X4_F32.
// 256-thread WG = 8 waves; wave w owns rows [wg*128 + w*16, +16) across all
// 128 columns (8 accumulator tiles). Both LDS tiles filled with
// global_load_async_to_lds_b128; one s_wait_asynccnt + barrier per K-block.
// ---------------------------------------------------------------------------
__global__ __launch_bounds__(THREADS)
void gemm_wmma_f32(const float* __restrict__ x,
                   const float* __restrict__ Wt,   // [HIDDEN][N_FEAT]
                   float* __restrict__ h) {
  __shared__ float sX [ROWS_PER_WG][KB + PAD];  // x rows tile
  __shared__ float sWt[HIDDEN]    [KB + PAD];   // W^T rows tile: sWt[n][k]

  const int tid  = threadIdx.x;
  const int wave = tid >> 5;
  const int lane = tid & 31;
  const int half = lane >> 4;   // 0: K pair 0,1 / M=v ; 1: K pair 2,3 / M=v+8
  const int l16  = lane & 15;
  const int row0 = blockIdx.x * ROWS_PER_WG;

  v8f acc[8];
#pragma unroll
  for (int t = 0; t < 8; ++t) acc[t] = (v8f){0.f,0.f,0.f,0.f,0.f,0.f,0.f,0.f};

  const int mrow = wave * 16 + l16;             // local A row for this lane

  for (int kb = 0; kb < N_FEAT; kb += KB) {
    // ---- async-stage x block: 128 rows x KB floats (1024 x 16B chunks) ----
#pragma unroll
    for (int i = 0; i < (ROWS_PER_WG * KB / 4) / THREADS; ++i) {
      int idx = i * THREADS + tid;
      int r   = idx / (KB / 4);
      int c4  = (idx % (KB / 4)) * 4;
      int gr  = row0 + r;
      if (gr > N_NODES - 1) gr = N_NODES - 1;   // clamp; tail rows masked at store
      async_cp16(&sX[r][c4], x + (size_t)gr * N_FEAT + kb + c4);
    }
    // ---- async-stage W^T block: 128 rows x KB floats ----
#pragma unroll
    for (int i = 0; i < (HIDDEN * KB / 4) / THREADS; ++i) {
      int idx = i * THREADS + tid;
      int r   = idx / (KB / 4);                 // n row of W^T
      int c4  = (idx % (KB / 4)) * 4;
      async_cp16(&sWt[r][c4], Wt + (size_t)r * N_FEAT + kb + c4);
    }
    wait_async0();
    __syncthreads();

    // ---- compute: 8 k-steps of K=4; batch loads, then 8 WMMAs ----
#pragma unroll
    for (int kk = 0; kk < KB; kk += 4) {
      // A frag: lanes 0-15 -> K=kk,kk+1 ; lanes 16-31 -> K=kk+2,kk+3
      v2f a = *(const v2f*)&sX[mrow][kk + 2 * half];
      v2f b[8];
#pragma unroll
      for (int t = 0; t < 8; ++t)
        b[t] = *(const v2f*)&sWt[t * 16 + l16][kk + 2 * half];
#pragma unroll
      for (int t = 0; t < 8; ++t)
        acc[t] = __builtin_amdgcn_wmma_f32_16x16x4_f32(
            /*neg_a=*/false, a, /*neg_b=*/false, b[t],
            /*c_mod=*/(short)0, acc[t], /*reuse_a=*/false, /*reuse_b=*/false);
    }
    __syncthreads();
  }

  // ---- store D tiles: VGPR v -> M = v + 8*half, N = l16 (per 16-col tile) ----
  const int myrow0 = row0 + wave * 16;
#pragma unroll
  for (int v = 0; v < 8; ++v) {
    int m = myrow0 + v + 8 * half;
    if (m < N_NODES) {
      float* hp = h + (size_t)m * HIDDEN + l16;
#pragma unroll
      for (int t = 0; t < 8; ++t) hp[t * 16] = acc[t][v];
    }
  }
}

// ---------------------------------------------------------------------------
// Kernel 2: out[r][n] = bias[n]  (vectorized float4; deterministic init)
// ---------------------------------------------------------------------------
__global__ void init_bias(const float* __restrict__ bias, float* __restrict__ out) {
  size_t i4   = (size_t)blockIdx.x * blockDim.x + threadIdx.x;
  size_t tot4 = (size_t)N_NODES * HIDDEN / 4;
  if (i4 < tot4) {
    int col4 = (int)(i4 % (HIDDEN / 4));
    float4 b = *(const float4*)(bias + col4 * 4);
    ((float4*)out)[i4] = b;
  }
}

// ---------------------------------------------------------------------------
// Kernel 3: COO SpMM scatter-add. One wave per edge; each lane handles 4 of
// 128 features. h and out (51.2 MB each) are L2-resident (192 MB L2): gathers
// hit L2, fp32 atomics execute at L2 (global_atomic_add_f32, non-returning).
// ---------------------------------------------------------------------------
__global__ __launch_bounds__(256)
void spmm_scatter(const int*   __restrict__ rows,
                  const int*   __restrict__ cols,
                  const float* __restrict__ vals,
                  const float* __restrict__ h,
                  float*       __restrict__ out,
                  int nEdges) {
  int e = blockIdx.x * (blockDim.x >> 5) + (threadIdx.x >> 5);
  if (e >= nEdges) return;
  int lane = threadIdx.x & 31;

  int   r = rows[e];
  int   c = cols[e];
  float v = vals[e];

  const float4 hv = *(const float4*)(h + (size_t)c * HIDDEN + lane * 4);
  float* op = out + (size_t)r * HIDDEN + lane * 4;
  unsafeAtomicAdd(op + 0, v * hv.x);
  unsafeAtomicAdd(op + 1, v * hv.y);
  unsafeAtomicAdd(op + 2, v * hv.z);
  unsafeAtomicAdd(op + 3, v * hv.w);
}

// ---------------------------------------------------------------------------
extern "C" void kernel_launch(void* const* d_in, const int* in_sizes, int n_in,
                              void* d_out, int out_size, void* d_ws, size_t ws_size,
                              hipStream_t stream) {
  const float* x    = (const float*)d_in[0];
  const int*   rows = (const int*)  d_in[1];
  const int*   cols = (const int*)  d_in[2];
  const float* vals = (const float*)d_in[3];
  const float* W    = (const float*)d_in[4];
  const float* bias = (const float*)d_in[5];
  float*       out  = (float*)d_out;

  float* h  = (float*)d_ws;                         // 100000*128*4 = 51.2 MB
  float* Wt = h + (size_t)N_NODES * HIDDEN;         // +256 KB
  const int nEdges = in_sizes[1];

  // 0) Wt = W^T (one-time, enables pure async-LDS staging in the GEMM)
  transpose_W<<<(N_FEAT * HIDDEN + 255) / 256, 256, 0, stream>>>(W, Wt);

  // 1) h = x @ W  (WMMA fp32, async-LDS staged tiles)
  int nWg = (N_NODES + ROWS_PER_WG - 1) / ROWS_PER_WG;
  gemm_wmma_f32<<<nWg, THREADS, 0, stream>>>(x, Wt, h);

  // 2) out = bias
  size_t tot4 = (size_t)N_NODES * HIDDEN / 4;
  init_bias<<<(int)((tot4 + 255) / 256), 256, 0, stream>>>(bias, out);

  // 3) out += segment_sum(vals * h[cols]) via L2 fp32 atomics
  int wavesPerBlk = 256 / 32;
  int nBlk = (nEdges + wavesPerBlk - 1) / wavesPerBlk;
  spmm_scatter<<<nBlk, 256, 0, stream>>>(rows, cols, vals, h, out, nEdges);
}